// DarcyResidual_38439957300077
// MI455X (gfx1250) — compile-verified
//
#include <hip/hip_runtime.h>
#include <stdint.h>

// Darcy residual: B=128, P=256. HBM-bandwidth-bound (~100MB traffic -> ~4.3us
// at 23.3 TB/s). CDNA5 path: TDM (tensor_load_to_lds) stages each
// (rows+halo)x256 tile of both channels into LDS (TENSORcnt + barrier), the
// stencil runs from LDS with per-thread rolling registers down the column
// (6 LDS dword reads per interior point instead of 10, keeping LDS bandwidth
// well under the 64-bank/WGP ceiling), non-temporal result stores.

#define PDIM  256
#define TROWS 32
#define MAXR  (TROWS + 2)

typedef uint32_t u32x4 __attribute__((ext_vector_type(4)));
typedef int      i32x4 __attribute__((ext_vector_type(4)));
typedef int      i32x8 __attribute__((ext_vector_type(8)));

#if defined(__HIP_DEVICE_COMPILE__) && __has_builtin(__builtin_amdgcn_tensor_load_to_lds)
#define USE_TDM 1
#else
#define USE_TDM 0
#endif

#if USE_TDM
// One Tensor-Data-Mover 2D tile load: rows x 256 f32, row stride 256 elements,
// packed contiguously into LDS at byte offset lds_off.
// Descriptor layout per CDNA5 ISA.
__device__ __forceinline__ void tdm_load_tile(const float* gsrc, uint32_t lds_off, int rows)
{
    uint64_t ga = (uint64_t)(uintptr_t)gsrc;
    u32x4 g0;
    g0[0] = 1u;                                    // count=1, user descriptor
    g0[1] = lds_off;                               // lds_addr (bytes, WG-relative)
    g0[2] = (uint32_t)ga;                          // global_addr[31:0]
    g0[3] = (uint32_t)((ga >> 32) & 0x01FFFFFFu)   // global_addr[56:32]
          | (2u << 30);                            // type = 2 ("image")
    i32x8 g1;
    g1[0] = (int)(2u << 16);                       // data_size = 4 bytes; wg_mask=0
    g1[1] = (int)(PDIM << 16);                     // tensor_dim0[15:0] = 256
    g1[2] = (int)((uint32_t)rows << 16);           // tensor_dim1[15:0] = rows
    g1[3] = (int)(PDIM << 16);                     // tile_dim0 = 256
    g1[4] = rows;                                  // tile_dim1 = rows; tile_dim2 = 0
    g1[5] = PDIM;                                  // tensor_dim0_stride = 256
    g1[6] = 0;                                     // stride0 hi / dim1_stride lo
    g1[7] = 0;                                     // dim1_stride hi
    i32x4 z4 = {0, 0, 0, 0};
#if __clang_major__ >= 23
    i32x8 z8 = {0, 0, 0, 0, 0, 0, 0, 0};
    __builtin_amdgcn_tensor_load_to_lds(g0, g1, z4, z4, z8, 0);
#else
    __builtin_amdgcn_tensor_load_to_lds(g0, g1, z4, z4, 0);
#endif
}
#endif

__global__ __launch_bounds__(256)
void DarcyResidual_38439957300077_kernel(const float* __restrict__ x,
                                         float* __restrict__ out)
{
    __shared__ float sA[MAXR * PDIM];
    __shared__ float sP[MAXR * PDIM];

    const int rb = blockIdx.x;    // row band 0..7
    const int b  = blockIdx.y;    // batch
    const int t  = threadIdx.x;   // 0..255 -> column

    const int r0  = rb * TROWS;
    const int rlo = (r0 == 0) ? 0 : (r0 - 1);
    int rhi = r0 + TROWS + 1;                 // exclusive
    if (rhi > PDIM) rhi = PDIM;
    const int nrows = rhi - rlo;

    const float* srcA = x + (size_t)b * 2u * (PDIM * PDIM) + (size_t)rlo * PDIM;
    const float* srcP = srcA + (size_t)(PDIM * PDIM);

#if USE_TDM
    if (t == 0) {
        // TDM issues once per executing wave (EXEC ignored); only wave 0 reaches here.
        tdm_load_tile(srcA, (uint32_t)(uintptr_t)&sA[0], nrows);
        tdm_load_tile(srcP, (uint32_t)(uintptr_t)&sP[0], nrows);
        __builtin_amdgcn_s_wait_tensorcnt(0);
    }
    __syncthreads();
#else
    // Fallback: tile is contiguous (tile width == row stride), coalesced copy.
    for (int idx = t; idx < nrows * PDIM; idx += 256) {
        sA[idx] = srcA[idx];
        sP[idx] = srcP[idx];
    }
    __syncthreads();
#endif

    const float inv2h = 128.0f;     // 1/(2H), H = 1/256
    const float invh2 = 65536.0f;   // 1/H^2
    const int   c     = t;

    float* outp = out + (size_t)b * (PDIM * PDIM);

    auto Av = [&](int r, int col) -> float {
        return (sA[r * PDIM + col] + 1.5f) * 5.0f;            // (x+1.5)/0.2
    };
    auto Pv = [&](int r, int col) -> float {
        return (sP[r * PDIM + col] + 0.9f) * (1.0f / 115.0f); // (x+0.9)/115
    };

    // Rolling registers down the column: rows g-1, g, g+1 (transformed values).
    const int lr0 = r0 - rlo;                  // local row of g = r0
    float pc = Pv(lr0, c),     ac = Av(lr0, c);
    float pp = Pv(lr0 + 1, c), ap = Av(lr0 + 1, c);            // row r0+1 always exists
    float pm = 0.0f, am = 0.0f;
    if (r0 > 0) { pm = Pv(lr0 - 1, c); am = Av(lr0 - 1, c); }  // unused when g==0

    for (int i = 0; i < TROWS; ++i) {
        const int g  = r0 + i;        // global row (uniform across block)
        const int lr = g - rlo;       // local row in LDS tile

        // ---- axis 0 (rows); branch is block-uniform ----
        float p_d0, p_d00, a_d0;
        if (g == 0) {
            // pp holds row 1 here.
            float p2 = Pv(2, c), p3 = Pv(3, c);
            p_d0  = (-3.0f * pc + 4.0f * pp - p2) * inv2h;
            p_d00 = ( 2.0f * pc - 5.0f * pp + 4.0f * p2 - p3) * invh2;
            a_d0  = (-3.0f * ac + 4.0f * ap - Av(2, c)) * inv2h;
        } else if (g == PDIM - 1) {
            // pm holds row 254 here.
            float q3 = Pv(lr - 2, c), q4 = Pv(lr - 3, c);
            p_d0  = ( 3.0f * pc - 4.0f * pm + q3) * inv2h;
            p_d00 = ( 2.0f * pc - 5.0f * pm + 4.0f * q3 - q4) * invh2;
            a_d0  = ( 3.0f * ac - 4.0f * am + Av(lr - 2, c)) * inv2h;
        } else {
            p_d0  = (pp - pm) * inv2h;
            p_d00 = (pp - 2.0f * pc + pm) * invh2;
            a_d0  = (ap - am) * inv2h;
        }

        // ---- axis 1 (cols); diverges only in edge waves ----
        float p_d1, p_d11, a_d1;
        if (c == 0) {
            float p1 = Pv(lr, 1), p2 = Pv(lr, 2), p3 = Pv(lr, 3);
            p_d1  = (-3.0f * pc + 4.0f * p1 - p2) * inv2h;
            p_d11 = ( 2.0f * pc - 5.0f * p1 + 4.0f * p2 - p3) * invh2;
            a_d1  = (-3.0f * ac + 4.0f * Av(lr, 1) - Av(lr, 2)) * inv2h;
        } else if (c == PDIM - 1) {
            float q2 = Pv(lr, c - 1), q3 = Pv(lr, c - 2), q4 = Pv(lr, c - 3);
            p_d1  = ( 3.0f * pc - 4.0f * q2 + q3) * inv2h;
            p_d11 = ( 2.0f * pc - 5.0f * q2 + 4.0f * q3 - q4) * invh2;
            a_d1  = ( 3.0f * ac - 4.0f * Av(lr, c - 1) + Av(lr, c - 2)) * inv2h;
        } else {
            float qm = Pv(lr, c - 1), qp = Pv(lr, c + 1);
            p_d1  = (qp - qm) * inv2h;
            p_d11 = (qp - 2.0f * pc + qm) * invh2;
            a_d1  = (Av(lr, c + 1) - Av(lr, c - 1)) * inv2h;
        }

        float res = -ac * (p_d00 + p_d11) - a_d0 * p_d0 - a_d1 * p_d1 - 1.0f;
        // Output is write-once, never re-read by this kernel: non-temporal store.
        __builtin_nontemporal_store(res, &outp[(size_t)g * PDIM + c]);

        // ---- rotate the row window; prefetch row g+2 for the next iteration ----
        if (i + 1 < TROWS) {
            pm = pc; pc = pp;
            am = ac; ac = ap;
            if (g + 2 < PDIM) {                 // next iteration's pp/ap (unused at g+1==255)
                pp = Pv(lr + 2, c);
                ap = Av(lr + 2, c);
            }
        }
    }
}

extern "C" void kernel_launch(void* const* d_in, const int* in_sizes, int n_in,
                              void* d_out, int out_size, void* d_ws, size_t ws_size,
                              hipStream_t stream) {
    const float* x   = (const float*)d_in[0];
    float*       out = (float*)d_out;
    const int    nb  = in_sizes[0] / (2 * PDIM * PDIM);   // batch = 128
    dim3 grid(PDIM / TROWS, nb);
    DarcyResidual_38439957300077_kernel<<<grid, 256, 0, stream>>>(x, out);
}